// QueryMixer_71691594105029
// MI455X (gfx1250) — compile-verified
//
#include <hip/hip_runtime.h>
#include <math.h>

// ---------------------------------------------------------------------------
// CDNA5 (gfx1250) fp32 WMMA: V_WMMA_F32_16X16X4_F32
//   A: 16x4 f32  -> 2 VGPRs/lane : A(m,k): lane = (k>=2)*16 + m, vgpr = k&1
//   B: 4x16 f32  -> 2 VGPRs/lane : B(k,n): lane = (k>=2)*16 + n, vgpr = k&1
//   C/D: 16x16 f32 -> 8 VGPRs    : vgpr r: row = r + 8*(lane>=16), col = lane&15
// ---------------------------------------------------------------------------
typedef __attribute__((ext_vector_type(2))) float v2f;
typedef __attribute__((ext_vector_type(8))) float v8f;

__device__ __forceinline__ v8f wmma_f32(v2f a, v2f b, v8f c) {
    return __builtin_amdgcn_wmma_f32_16x16x4_f32(
        /*neg_a=*/false, a, /*neg_b=*/false, b,
        /*c_mod=*/(short)0, c, /*reuse_a=*/false, /*reuse_b=*/false);
}

// ---------------------------------------------------------------------------
// Tensor Data Mover (TDM): tensor_load_to_lds, 2-D tile with HW LDS padding
// ---------------------------------------------------------------------------
#if defined(__AMDGCN__) && __has_builtin(__builtin_amdgcn_tensor_load_to_lds) && \
    __has_builtin(__builtin_amdgcn_s_wait_tensorcnt)
#define HAVE_TDM 1
#else
#define HAVE_TDM 0
#endif

#if HAVE_TDM
typedef unsigned int u32x4 __attribute__((ext_vector_type(4)));
typedef int          i32x4 __attribute__((ext_vector_type(4)));
typedef int          i32x8 __attribute__((ext_vector_type(8)));

// D# per ISA ch.8: group0 = {count/lds_addr/global_addr/type},
// group1 = {flags, tensor_dim0/1, tile_dim0/1, tensor_dim0_stride}.
// data_size = 4 B elements.  pad_interval code: pad after 2^(c+1) DWORDs.
// Rows past tensor_dim1 are zero-filled by HW (used for the 300->304 K pad).
__device__ __forceinline__ void tdm_load_tile_2d(
    const void* lds_dst, const float* gsrc,
    unsigned tensor_d0, unsigned tensor_d1, unsigned row_stride_elems,
    unsigned tile_d0, unsigned tile_d1,
    unsigned pad_interval_code, unsigned pad_amount_code)
{
    const unsigned long long ga = (unsigned long long)(size_t)gsrc;
    const unsigned lds_addr = (unsigned)(size_t)lds_dst;

    u32x4 g0;
    g0[0] = 1u;                                          // count=1 (valid)
    g0[1] = lds_addr;                                    // LDS byte address
    g0[2] = (unsigned)ga;                                // global_addr[31:0]
    g0[3] = (unsigned)((ga >> 32) & 0x01ffffffu) | (2u << 30);   // [56:32] | type=2

    i32x8 g1;
    g1[0] = (int)((2u << 16)                             // data_size = 4 B
                | (1u << 20)                             // pad_enable
                | (pad_interval_code << 22)
                | (pad_amount_code << 25));
    g1[1] = (int)((tensor_d0 & 0xffffu) << 16);          // abar=0 | td0[15:0]
    g1[2] = (int)(((tensor_d0 >> 16) & 0xffffu) | ((tensor_d1 & 0xffffu) << 16));
    g1[3] = (int)(((tensor_d1 >> 16) & 0xffffu) | ((tile_d0 & 0xffffu) << 16));
    g1[4] = (int)(tile_d1 & 0xffffu);                    // tile_dim1 | tile_dim2=0
    g1[5] = (int)row_stride_elems;                       // dim0_stride[31:0]
    g1[6] = 0;                                           // stride[47:32] | d1s[15:0]
    g1[7] = 0;

    const i32x4 z4 = {0, 0, 0, 0};
#if __clang_major__ >= 23
    const i32x8 z8 = {0, 0, 0, 0, 0, 0, 0, 0};
    __builtin_amdgcn_tensor_load_to_lds(g0, g1, z4, z4, z8, 0);
#else
    __builtin_amdgcn_tensor_load_to_lds(g0, g1, z4, z4, 0);
#endif
}
#endif  // HAVE_TDM

// ---------------------------------------------------------------------------
// Problem constants (B,Q,D,H,EF = 64,300,512,8,4)
// ---------------------------------------------------------------------------
#define NB   64
#define NQ   300
#define ND   512
#define NH   8
#define NDH  64
#define NDV  256
#define NDEF 2048      // D*EF
#define NM   19200     // B*Q
#define QT16 19        // ceil(300/16)
#define KPAD 304       // 300 padded to multiple of 4 (and 16)
#define SLD  305       // odd row stride for score tile (bank-conflict free)

// ===========================================================================
// Kernel 1/3: C[M,N] = A[M,K] @ W[K,N] + bias[N] (+ residual[M,N])
// M % 64 == 0, N % 64 == 0, K % 16 == 0.  256 threads, 64x64 tile/WG.
// TDM path: double-buffered LDS tiles DMA'd by wave 0, overlapped with WMMA.
// ===========================================================================
__global__ __launch_bounds__(256)
void gemm_bias_kernel(const float* __restrict__ A, const float* __restrict__ W,
                      const float* __restrict__ bias,
                      const float* __restrict__ residual,
                      float* __restrict__ C, int N, int K)
{
    __shared__ float As[2][64][17];   // 64 x 16 tile, padded stride 17
    __shared__ float Bs[2][16][65];   // 16 x 64 tile, padded stride 65

    const int tid  = threadIdx.x;
    const int wave = tid >> 5;
    const int lane = tid & 31;
    const int g    = lane >> 4;    // K half-group select
    const int ln   = lane & 15;

    const int rowBlk = blockIdx.x * 64;
    const int colBlk = blockIdx.y * 64;

    // two 16x16 subtiles per wave sharing the A fragment
    const int sm  = wave >> 1;          // 0..3
    const int sn0 = (wave & 1) * 2;     // 0 or 2

    v8f acc0 = {};
    v8f acc1 = {};

    const int steps = K >> 4;

#if HAVE_TDM
    if (tid == 0) {  // TDM ignores EXEC; one wave issues the whole-tile DMA
        tdm_load_tile_2d(&As[0][0][0], A + (size_t)rowBlk * K,
                         16, 64, (unsigned)K, 16, 64, /*pad 16 dw*/3, 0);
        tdm_load_tile_2d(&Bs[0][0][0], W + colBlk,
                         64, 16, (unsigned)N, 64, 16, /*pad 64 dw*/5, 0);
    }
#else
    const int arow = tid >> 2, acol = (tid & 3) * 4;     // 64 rows x 16 cols
    const int brow = tid >> 4, bcol = (tid & 15) * 4;    // 16 rows x 64 cols
#endif

    for (int i = 0; i < steps; ++i) {
#if HAVE_TDM
        __syncthreads();                 // all readers of buf[(i+1)&1] done
        if (tid == 0) {
            if (i + 1 < steps) {
                const int k1 = (i + 1) << 4;
                tdm_load_tile_2d(&As[(i + 1) & 1][0][0],
                                 A + (size_t)rowBlk * K + k1,
                                 16, 64, (unsigned)K, 16, 64, 3, 0);
                tdm_load_tile_2d(&Bs[(i + 1) & 1][0][0],
                                 W + (size_t)k1 * N + colBlk,
                                 64, 16, (unsigned)N, 64, 16, 5, 0);
                __builtin_amdgcn_s_wait_tensorcnt(2);  // pair(i) done
            } else {
                __builtin_amdgcn_s_wait_tensorcnt(0);
            }
        }
        __syncthreads();                 // buf[i&1] visible to all waves
        const int buf = i & 1;
#else
        const int k0 = i << 4;
        const float4 av = *(const float4*)(A + (size_t)(rowBlk + arow) * K + k0 + acol);
        const float4 bv = *(const float4*)(W + (size_t)(k0 + brow) * N + colBlk + bcol);
        if (k0 + 16 < K) {
            __builtin_prefetch(A + (size_t)(rowBlk + arow) * K + k0 + 16 + acol, 0, 0);
            __builtin_prefetch(W + (size_t)(k0 + 16 + brow) * N + colBlk + bcol, 0, 0);
        }
        __syncthreads();
        As[0][arow][acol + 0] = av.x; As[0][arow][acol + 1] = av.y;
        As[0][arow][acol + 2] = av.z; As[0][arow][acol + 3] = av.w;
        Bs[0][brow][bcol + 0] = bv.x; Bs[0][brow][bcol + 1] = bv.y;
        Bs[0][brow][bcol + 2] = bv.z; Bs[0][brow][bcol + 3] = bv.w;
        __syncthreads();
        const int buf = 0;
#endif

#pragma unroll
        for (int kk = 0; kk < 4; ++kk) {
            const int ka = kk * 4 + 2 * g;
            v2f a;  a.x  = As[buf][sm * 16 + ln][ka];
                    a.y  = As[buf][sm * 16 + ln][ka + 1];
            v2f b0; b0.x = Bs[buf][ka][sn0 * 16 + ln];
                    b0.y = Bs[buf][ka + 1][sn0 * 16 + ln];
            v2f b1; b1.x = Bs[buf][ka][(sn0 + 1) * 16 + ln];
                    b1.y = Bs[buf][ka + 1][(sn0 + 1) * 16 + ln];
            acc0 = wmma_f32(a, b0, acc0);
            acc1 = wmma_f32(a, b1, acc1);
        }
    }

    const int rowBase = rowBlk + sm * 16 + g * 8;
    const int c0 = colBlk + sn0 * 16 + ln;
    const int c1 = c0 + 16;
    const float bia0 = bias[c0], bia1 = bias[c1];
#pragma unroll
    for (int i = 0; i < 8; ++i) {
        const int r = rowBase + i;
        float o0 = acc0[i] + bia0;
        float o1 = acc1[i] + bia1;
        if (residual) {
            o0 += residual[(size_t)r * N + c0];
            o1 += residual[(size_t)r * N + c1];
        }
        C[(size_t)r * N + c0] = o0;
        C[(size_t)r * N + c1] = o1;
    }
}

// ===========================================================================
// Kernel 2: fused attention for one (b, h, q-tile of 16 rows)
//   S = gelu(q @ M * wb + bias); A = softmax(S); O = A @ V
//   grid = (B*H, 19), block = 256 (8 waves)
// Dynamic LDS layout:
//   S    [16][SLD]        score/attn tile (fp32)
//   Aqs  [16][65]         q tile  (16 x 64)
//   Bms  [KPAD][65]       mixing tile M^T stored as [kcol][dh]
// TDM path: both tiles DMA'd; rows >= 300 zero-filled by the tensor engine.
// ===========================================================================
__global__ __launch_bounds__(256)
void attn_kernel(const float* __restrict__ qf,    // [B*Q, 512]
                 const float* __restrict__ G,     // [B*Q, 512]  gen projection
                 const float* __restrict__ V,     // [B*Q, 2048] val projection
                 const float* __restrict__ wb,    // [H, Q, Q]
                 const float* __restrict__ bias,  // [H, Q, Q]
                 float* __restrict__ O)           // [B*Q, 2048]
{
    extern __shared__ float smem[];
    float* S   = smem;                  // 16 * SLD
    float* Aqs = S + 16 * SLD;          // 16 * 65
    float* Bms = Aqs + 16 * 65;         // KPAD * 65

    const int bh = blockIdx.x;
    const int qt = blockIdx.y;
    const int b  = bh / NH;
    const int h  = bh % NH;
    const int qbase = qt * 16;

    const int tid  = threadIdx.x;
    const int wave = tid >> 5;
    const int lane = tid & 31;
    const int g    = lane >> 4;
    const int ln   = lane & 15;

#if HAVE_TDM
    if (tid == 0) {
        // q tile: 16 rows x 64; rows past NQ zero-filled (tensor_dim1 clips)
        tdm_load_tile_2d(Aqs, qf + (size_t)(b * NQ + qbase) * ND + h * NDH,
                         64, (unsigned)(NQ - qbase), ND, 64, 16, 5, 0);
        // mixing tile: 304 rows x 64; rows 300..303 zero-filled by HW
        tdm_load_tile_2d(Bms, G + (size_t)(b * NQ) * ND + h * NDH,
                         64, NQ, ND, 64, KPAD, 5, 0);
        __builtin_amdgcn_s_wait_tensorcnt(0);
    }
#else
    {
        const int r  = tid >> 4;
        const int c4 = (tid & 15) * 4;
        const int qr = min(qbase + r, NQ - 1);
        const float4 v = *(const float4*)(qf + (size_t)(b * NQ + qr) * ND + h * NDH + c4);
        Aqs[r * 65 + c4 + 0] = v.x; Aqs[r * 65 + c4 + 1] = v.y;
        Aqs[r * 65 + c4 + 2] = v.z; Aqs[r * 65 + c4 + 3] = v.w;
#pragma unroll
        for (int it = 0; it < QT16; ++it) {
            const int rr = it * 16 + r;            // 0..303
            const int kr = min(rr, NQ - 1);
            const float4 w = *(const float4*)(G + (size_t)(b * NQ + kr) * ND + h * NDH + c4);
            Bms[rr * 65 + c4 + 0] = w.x; Bms[rr * 65 + c4 + 1] = w.y;
            Bms[rr * 65 + c4 + 2] = w.z; Bms[rr * 65 + c4 + 3] = w.w;
        }
    }
#endif
    __syncthreads();

    // ---- stage 1: scores[16 x 304] = q[16x64] @ M[64x304], epilogue fused ----
    for (int nt = wave; nt < QT16; nt += 8) {
        v8f acc = {};
#pragma unroll
        for (int kk = 0; kk < 16; ++kk) {          // 64 / 4
            const int ka = kk * 4 + 2 * g;
            v2f a;  a.x = Aqs[ln * 65 + ka];              a.y = Aqs[ln * 65 + ka + 1];
            v2f bf; bf.x = Bms[(nt * 16 + ln) * 65 + ka]; bf.y = Bms[(nt * 16 + ln) * 65 + ka + 1];
            acc = wmma_f32(a, bf, acc);
        }
        const int kcol = nt * 16 + ln;
#pragma unroll
        for (int i = 0; i < 8; ++i) {
            const int rloc = i + 8 * g;
            float sv = 0.0f;
            if (kcol < NQ) {
                const int qr = min(qbase + rloc, NQ - 1);
                const size_t idx = ((size_t)h * NQ + qr) * NQ + kcol;
                const float x = acc[i] * wb[idx] + bias[idx];
                sv = 0.5f * x * (1.0f + erff(x * 0.70710678118654752f));  // exact GELU
            }
            S[rloc * SLD + kcol] = sv;   // pad cols 300..303 get exact 0
        }
    }
    __syncthreads();

    // ---- softmax over k (300 valid cols); 2 rows per wave, 16 lanes per row ----
    {
        const int row = wave * 2 + g;
        float* Srow = S + row * SLD;
        float mx = -3.0e38f;
        for (int k = ln; k < NQ; k += 16) mx = fmaxf(mx, Srow[k]);
#pragma unroll
        for (int o = 8; o; o >>= 1) mx = fmaxf(mx, __shfl_xor(mx, o, 32));
        float sum = 0.0f;
        for (int k = ln; k < NQ; k += 16) {
            const float e = expf(Srow[k] - mx);
            Srow[k] = e;
            sum += e;
        }
#pragma unroll
        for (int o = 8; o; o >>= 1) sum += __shfl_xor(sum, o, 32);
        const float inv = 1.0f / sum;
        for (int k = ln; k < NQ; k += 16) Srow[k] *= inv;
    }
    __syncthreads();

    // ---- stage 2: O[16 x 256] = attn[16 x 304] @ V[304 x 256] ----
#pragma unroll
    for (int half = 0; half < 2; ++half) {
        const int nt = wave + half * 8;            // 0..15
        v8f acc = {};
        const float* Vcol = V + (size_t)(b * NQ) * NDEF + h * NDV + nt * 16 + ln;
        for (int kk = 0; kk < KPAD / 4; ++kk) {    // 76 chunks of 4
            const int ka = kk * 4 + 2 * g;
            v2f a;  a.x = S[ln * SLD + ka];        a.y = S[ln * SLD + ka + 1];
            const int k0r = min(ka, NQ - 1);       // clamped; attn is 0 beyond 299
            const int k1r = min(ka + 1, NQ - 1);
            v2f bf;
            bf.x = Vcol[(size_t)k0r * NDEF];
            bf.y = Vcol[(size_t)k1r * NDEF];
            acc = wmma_f32(a, bf, acc);
        }
        const int ocol = h * NDV + nt * 16 + ln;
#pragma unroll
        for (int i = 0; i < 8; ++i) {
            const int qr = qbase + i + 8 * g;
            if (qr < NQ)
                O[(size_t)(b * NQ + qr) * NDEF + ocol] = acc[i];
        }
    }
}

// ===========================================================================
// Launch: GEMM(gen) -> GEMM(val) -> fused attention -> GEMM(out, +residual)
// Workspace: G [19200x512] | V [19200x2048] | OA [19200x2048]  (~354 MB fp32)
// ===========================================================================
extern "C" void kernel_launch(void* const* d_in, const int* in_sizes, int n_in,
                              void* d_out, int out_size, void* d_ws, size_t ws_size,
                              hipStream_t stream) {
    (void)in_sizes; (void)n_in; (void)out_size; (void)ws_size;

    const float* qf    = (const float*)d_in[0];
    const float* bias  = (const float*)d_in[1];
    const float* W_gen = (const float*)d_in[2];
    const float* b_gen = (const float*)d_in[3];
    const float* W_val = (const float*)d_in[4];
    const float* b_val = (const float*)d_in[5];
    const float* W_out = (const float*)d_in[6];
    const float* b_out = (const float*)d_in[7];
    const float* wbank = (const float*)d_in[8];
    float* out = (float*)d_out;

    float* G  = (float*)d_ws;                       // [19200 x 512]
    float* V  = G + (size_t)NM * ND;                // [19200 x 2048]
    float* OA = V + (size_t)NM * NDEF;              // [19200 x 2048]

    const dim3 blk(256);

    // G = qf @ W_gen + b_gen
    gemm_bias_kernel<<<dim3(NM / 64, ND / 64), blk, 0, stream>>>(
        qf, W_gen, b_gen, nullptr, G, ND, ND);
    // V = qf @ W_val + b_val
    gemm_bias_kernel<<<dim3(NM / 64, NDEF / 64), blk, 0, stream>>>(
        qf, W_val, b_val, nullptr, V, NDEF, ND);
    // fused attention
    const size_t smem_bytes = (size_t)(16 * SLD + 16 * 65 + KPAD * 65) * sizeof(float);
    attn_kernel<<<dim3(NB * NH, QT16), blk, smem_bytes, stream>>>(
        qf, G, V, wbank, bias, OA);
    // out = OA @ W_out + b_out + qf
    gemm_bias_kernel<<<dim3(NM / 64, ND / 64), blk, 0, stream>>>(
        OA, W_out, b_out, qf, out, ND, NDEF);
}